// CascadeGDCN_17162689315367
// MI455X (gfx1250) — compile-verified
//
#include <hip/hip_runtime.h>
#include <hip/hip_bf16.h>

#define N_NODES 100000
#define N_EDGES 1600000
#define D 64
#define NF (N_NODES * D)

typedef float v2f __attribute__((ext_vector_type(2)));
typedef float v8f __attribute__((ext_vector_type(8)));

// ---------------------------------------------------------------------------
// X_out0 = max(out_degree,1e-8) * H ; X_in0 = max(in_degree,1e-8) * H
// One thread per 4 features (float4).
// ---------------------------------------------------------------------------
__global__ void __launch_bounds__(256) scale_kernel(
    const float* __restrict__ H,
    const float* __restrict__ outdeg,
    const float* __restrict__ indeg,
    float* __restrict__ Xout,
    float* __restrict__ Xin) {
  int t = blockIdx.x * blockDim.x + threadIdx.x;
  if (t >= NF / 4) return;
  int node = t / (D / 4);
  float doo = fmaxf(outdeg[node], 1e-8f);
  float dii = fmaxf(indeg[node], 1e-8f);
  float4 h = reinterpret_cast<const float4*>(H)[t];
  float4 o, i;
  o.x = doo * h.x; o.y = doo * h.y; o.z = doo * h.z; o.w = doo * h.w;
  i.x = dii * h.x; i.y = dii * h.y; i.z = dii * h.z; i.w = dii * h.w;
  reinterpret_cast<float4*>(Xout)[t] = o;
  reinterpret_cast<float4*>(Xin)[t]  = i;
}

// ---------------------------------------------------------------------------
// dst[didx[e], :] += val[e] * src[sidx[e], :]
// One thread per (edge, 4-feature chunk): 16 threads per edge.
// Gather is a coalesced float4 (b128) load; scatter uses relaxed agent-scope
// FP32 atomics -> native no-return global_atomic_add_f32 (no CAS loop).
// ---------------------------------------------------------------------------
__device__ __forceinline__ void atomic_add_f32(float* p, float v) {
  (void)__hip_atomic_fetch_add(p, v, __ATOMIC_RELAXED, __HIP_MEMORY_SCOPE_AGENT);
}

__global__ void __launch_bounds__(256) scatter_kernel(
    float* __restrict__ dst,
    const float* __restrict__ src,
    const int* __restrict__ didx,
    const int* __restrict__ sidx,
    const float* __restrict__ val) {
  long long t = (long long)blockIdx.x * blockDim.x + threadIdx.x;
  long long e = t >> 4;
  int q = (int)(t & 15);
  if (e >= N_EDGES) return;
  int r = didx[e];
  int c = sidx[e];
  float v = val[e];
  float4 x = reinterpret_cast<const float4*>(src + (size_t)c * D)[q];
  float* dp = dst + (size_t)r * D + (size_t)q * 4;
  atomic_add_f32(dp + 0, v * x.x);
  atomic_add_f32(dp + 1, v * x.y);
  atomic_add_f32(dp + 2, v * x.z);
  atomic_add_f32(dp + 3, v * x.w);
}

// ---------------------------------------------------------------------------
// sum += alpha[k]*theta_out[k]*Xout + alpha[k]*theta_in[k]*Xin
// alpha = softmax(hop_attention) recomputed per thread (3 cached scalars).
// ---------------------------------------------------------------------------
__global__ void __launch_bounds__(256) accum_kernel(
    float* __restrict__ sum,
    const float* __restrict__ Xout,
    const float* __restrict__ Xin,
    const float* __restrict__ hop,
    const float* __restrict__ th_out,
    const float* __restrict__ th_in,
    int k) {
  int t = blockIdx.x * blockDim.x + threadIdx.x;
  if (t >= NF / 4) return;
  float m = fmaxf(fmaxf(hop[0], hop[1]), hop[2]);
  float e0 = __expf(hop[0] - m);
  float e1 = __expf(hop[1] - m);
  float e2 = __expf(hop[2] - m);
  float a = ((k == 0) ? e0 : (k == 1) ? e1 : e2) / (e0 + e1 + e2);
  float co = a * th_out[k];
  float ci = a * th_in[k];
  float4 s  = reinterpret_cast<const float4*>(sum)[t];
  float4 xo = reinterpret_cast<const float4*>(Xout)[t];
  float4 xi = reinterpret_cast<const float4*>(Xin)[t];
  s.x += co * xo.x + ci * xi.x;
  s.y += co * xo.y + ci * xi.y;
  s.z += co * xo.z + ci * xi.z;
  s.w += co * xo.w + ci * xi.w;
  reinterpret_cast<float4*>(sum)[t] = s;
}

// ---------------------------------------------------------------------------
// out = sigmoid(S @ Theta) + H, fused, in place on S (== d_out).
// One wave32 per 16-row tile of S. 4 accumulators (16x16 each) cover N=64.
// K=64 accumulated via 16x V_WMMA_F32_16X16X4_F32 per accumulator.
//
// f32 WMMA fragment layouts (ISA 7.12.2):
//   A 16x4:  lane l holds A[l%16][2*(l/16) + {0,1}]           (float2)
//   B 4x16:  lane l holds B[2*(l/16) + {0,1}][l%16]           (float2)
//   C/D:     VGPR v of lane l holds D[v + 8*(l/16)][l%16]     (float8)
// All S loads occur before any S store (same pointer -> order preserved).
// ---------------------------------------------------------------------------
__global__ void __launch_bounds__(256) gemm_sig_add_kernel(
    float* S,                         // sum_term in, result out (d_out)
    const float* __restrict__ Theta,  // 64x64 row-major (K x N)
    const float* __restrict__ H) {
  int lane = threadIdx.x & 31;
  int tile = blockIdx.x * (blockDim.x >> 5) + (threadIdx.x >> 5);
  if (tile >= N_NODES / 16) return;

  int m0 = tile * 16;
  int mrow  = lane & 15;
  int khalf = lane >> 4;

  const float* arow  = S + (size_t)(m0 + mrow) * D + 2 * khalf;
  const float* bbase = Theta + (size_t)(2 * khalf) * D + (lane & 15);

  v8f c0 = {}, c1 = {}, c2 = {}, c3 = {};
#pragma unroll
  for (int kk = 0; kk < 16; ++kk) {
    v2f a;
    a.x = arow[4 * kk];
    a.y = arow[4 * kk + 1];
    const float* bp = bbase + (size_t)(4 * kk) * D;
    v2f b0, b1, b2, b3;
    b0.x = bp[0];      b0.y = bp[D];
    b1.x = bp[16];     b1.y = bp[16 + D];
    b2.x = bp[32];     b2.y = bp[32 + D];
    b3.x = bp[48];     b3.y = bp[48 + D];
    c0 = __builtin_amdgcn_wmma_f32_16x16x4_f32(false, a, false, b0, (short)0, c0, false, false);
    c1 = __builtin_amdgcn_wmma_f32_16x16x4_f32(false, a, false, b1, (short)0, c1, false, false);
    c2 = __builtin_amdgcn_wmma_f32_16x16x4_f32(false, a, false, b2, (short)0, c2, false, false);
    c3 = __builtin_amdgcn_wmma_f32_16x16x4_f32(false, a, false, b3, (short)0, c3, false, false);
  }

  int colb = lane & 15;
#pragma unroll
  for (int v = 0; v < 8; ++v) {
    size_t rb = (size_t)(m0 + v + 8 * khalf) * D;
    float x0 = c0[v], x1 = c1[v], x2 = c2[v], x3 = c3[v];
    S[rb +  0 + colb] = 1.0f / (1.0f + __expf(-x0)) + H[rb +  0 + colb];
    S[rb + 16 + colb] = 1.0f / (1.0f + __expf(-x1)) + H[rb + 16 + colb];
    S[rb + 32 + colb] = 1.0f / (1.0f + __expf(-x2)) + H[rb + 32 + colb];
    S[rb + 48 + colb] = 1.0f / (1.0f + __expf(-x3)) + H[rb + 48 + colb];
  }
}

// ---------------------------------------------------------------------------
extern "C" void kernel_launch(void* const* d_in, const int* in_sizes, int n_in,
                              void* d_out, int out_size, void* d_ws, size_t ws_size,
                              hipStream_t stream) {
  const float* H      = (const float*)d_in[0];
  const int*   erow   = (const int*)d_in[1];
  const int*   ecol   = (const int*)d_in[2];
  const float* eval   = (const float*)d_in[3];
  const float* outdeg = (const float*)d_in[4];
  const float* indeg  = (const float*)d_in[5];
  const float* hop    = (const float*)d_in[6];
  const float* th_out = (const float*)d_in[7];
  const float* th_in  = (const float*)d_in[8];
  const float* Theta  = (const float*)d_in[9];

  float* sum = (float*)d_out;           // sum_term accumulates in d_out
  float* ws  = (float*)d_ws;
  float* Xo[2] = {ws,              ws + (size_t)NF};
  float* Xi[2] = {ws + 2ull * NF,  ws + 3ull * NF};

  const int elem_blocks = (NF / 4 + 255) / 256;

  hipMemsetAsync(sum, 0, sizeof(float) * (size_t)NF, stream);
  scale_kernel<<<elem_blocks, 256, 0, stream>>>(H, outdeg, indeg, Xo[0], Xi[0]);

  int cur = 0;
  for (int k = 0; k < 3; ++k) {
    int nxt = cur ^ 1;
    hipMemsetAsync(Xo[nxt], 0, sizeof(float) * (size_t)NF, stream);
    hipMemsetAsync(Xi[nxt], 0, sizeof(float) * (size_t)NF, stream);

    long long nthreads = (long long)N_EDGES * 16;
    int sblocks = (int)((nthreads + 255) / 256);
    // X_out = A @ X_out  (edge (r,c): dst r, src c)
    scatter_kernel<<<sblocks, 256, 0, stream>>>(Xo[nxt], Xo[cur], erow, ecol, eval);
    // X_in = A^T @ X_in  (edge (r,c): dst c, src r)
    scatter_kernel<<<sblocks, 256, 0, stream>>>(Xi[nxt], Xi[cur], ecol, erow, eval);

    accum_kernel<<<elem_blocks, 256, 0, stream>>>(sum, Xo[nxt], Xi[nxt],
                                                  hop, th_out, th_in, k);
    cur = nxt;
  }

  const int tiles = N_NODES / 16;               // 6250
  gemm_sig_add_kernel<<<(tiles + 7) / 8, 256, 0, stream>>>(sum, Theta, H);
}